// MyModel_87454124082174
// MI455X (gfx1250) — compile-verified
//
#include <hip/hip_runtime.h>
#include <stdint.h>

#define NBOX   20000
#define NCLS   80
#define NB     8
#define KCAND  256
#define GBUF   1024
#define MAXPC  8

typedef uint32_t v4u __attribute__((ext_vector_type(4)));
typedef uint32_t v8u __attribute__((ext_vector_type(8)));

// ---------------------------------------------------------------------------
// Kernel A: one workgroup per (batch, class).
//  1) TDM tensor_load_to_lds: DMA the 20000-row x 1-col score column
//     (row stride 80 floats) straight into LDS (single descriptor, wave 0)
//  2) exact top-256 via 3-pass radix select + 64-bit-key bitonic sort
//  3) async-DMA (per-lane gather) the 256 candidate boxes into LDS
//  4) 256x256 IoU suppression bit-matrix, greedy keep (max 8)
//  5) write kept slots {score, ti, box} to workspace
// ---------------------------------------------------------------------------
__global__ __launch_bounds__(256) void nms_class_kernel(
    const float* __restrict__ boxes, const float* __restrict__ scores,
    float* __restrict__ slot_score, int* __restrict__ slot_ti,
    float4* __restrict__ slot_box)
{
  const int c   = blockIdx.x;   // class 0..79
  const int b   = blockIdx.y;   // batch 0..7
  const int tid = threadIdx.x;  // 0..255

  __shared__ uint32_t           s_bits[NBOX];      // 80 KB: score bit patterns
  __shared__ uint32_t           s_hist[256];
  __shared__ unsigned long long s_keys[GBUF];      // 8 KB: (scorebits<<32)|~idx
  __shared__ float4             s_box[KCAND];      // 4 KB
  __shared__ uint32_t           s_sup[KCAND * 8];  // 8 KB suppression bitmatrix
  __shared__ uint32_t           s_cnt, s_prefix, s_need;

  // ---- 1) stage class-c score column into LDS via the Tensor Data Mover ----
  // Tensor: 20000 rows x 80 cols f32, row stride 80.  Tile: 1 x 20000 column.
  // One TDM descriptor replaces ~2500 per-lane async loads; the DMA engine
  // walks the strided column while wave 0 waits on TENSORcnt.
  if (tid < 32) {  // wave 0 only: TDM ignores EXEC, avoid 8x duplicate DMA
    uint64_t gaddr = (uint64_t)(uintptr_t)(scores + ((size_t)b * NBOX * NCLS + c));
    uint32_t laddr = (uint32_t)(uintptr_t)(&s_bits[0]);

    v4u g0;
    g0.x = 1u;                                   // count=1, user desc, no gather
    g0.y = laddr;                                // lds_addr (bytes)
    g0.z = (uint32_t)(gaddr & 0xFFFFFFFFull);    // global_addr[31:0]
    g0.w = (uint32_t)((gaddr >> 32) & 0x01FFFFFFull) | 0x80000000u; // [56:32]|type=2

    v8u g1;
    g1.s0 = 2u << 16;                            // wg_mask=0, data_size=4B
    g1.s1 = ((uint32_t)NCLS & 0xFFFFu) << 16;    // tensor_dim0 = 80 (lo16)
    g1.s2 = ((uint32_t)NBOX & 0xFFFFu) << 16;    // dim0 hi=0 | tensor_dim1 lo16
    g1.s3 = 1u << 16;                            // dim1 hi=0 | tile_dim0 = 1
    g1.s4 = (uint32_t)NBOX;                      // tile_dim1 = 20000, tile_dim2=0
    g1.s5 = (uint32_t)NCLS;                      // tensor_dim0_stride = 80
    g1.s6 = 0u;                                  // stride hi | dim1_stride lo
    g1.s7 = 0u;                                  // dim1_stride hi

    asm volatile("tensor_load_to_lds %0, %1" :: "s"(g0), "s"(g1) : "memory");
    __builtin_amdgcn_s_wait_tensorcnt(0);
  }
  __syncthreads();

  // ---- 2a) radix select: find 24-bit prefix of the 256th largest score ----
  uint32_t prefix = 0, need = KCAND;
  for (int pass = 0; pass < 3; ++pass) {
    const int shift = 24 - pass * 8;
    s_hist[tid] = 0;
    __syncthreads();
    for (int i = tid; i < NBOX; i += 256) {
      uint32_t v = s_bits[i];
      if ((pass == 0) || ((v >> (shift + 8)) == prefix))
        atomicAdd(&s_hist[(v >> shift) & 0xFFu], 1u);
    }
    __syncthreads();
    if (tid == 0) {
      uint32_t nd = need;
      int bin = 0;
      for (int q = 255; q >= 0; --q) {
        uint32_t cc = s_hist[q];
        if (nd <= cc) { bin = q; break; }
        nd -= cc;
      }
      s_prefix = (prefix << 8) | (uint32_t)bin;
      s_need   = nd;
    }
    __syncthreads();
    prefix = s_prefix;
    need   = s_need;
    __syncthreads();
  }

  // ---- 2b) gather all candidates with 24-bit prefix >= pivot ----
  if (tid == 0) s_cnt = 0;
  __syncthreads();
  for (int i = tid; i < NBOX; i += 256) {
    uint32_t v = s_bits[i];
    if ((v >> 8) >= prefix) {
      uint32_t pos = atomicAdd(&s_cnt, 1u);
      if (pos < GBUF)
        s_keys[pos] = ((unsigned long long)v << 32) | (uint32_t)(~(uint32_t)i);
    }
  }
  __syncthreads();
  uint32_t n = s_cnt < GBUF ? s_cnt : GBUF;
  for (int i = tid; i < GBUF; i += 256)
    if ((uint32_t)i >= n) s_keys[i] = 0ull;
  __syncthreads();

  // ---- 2c) bitonic sort descending (ties -> smaller original index first) --
  for (unsigned k = 2; k <= GBUF; k <<= 1) {
    for (unsigned j = k >> 1; j > 0; j >>= 1) {
      for (unsigned zb = 0; zb < GBUF; zb += 256) {
        unsigned i   = zb + tid;
        unsigned ixj = i ^ j;
        if (ixj > i) {
          unsigned long long a = s_keys[i], d = s_keys[ixj];
          bool up = ((i & k) == 0);
          if (up ? (a < d) : (a > d)) { s_keys[i] = d; s_keys[ixj] = a; }
        }
      }
      __syncthreads();
    }
  }

  // ---- 3) per-lane async gather of the top-256 boxes into LDS ----
  {
    unsigned long long key = s_keys[tid];
    uint32_t idx = (key != 0ull) ? (uint32_t)(~((uint32_t)key)) : 0u;
    if (idx >= NBOX) idx = 0;
    uint64_t bbase = (uint64_t)(uintptr_t)(boxes + (size_t)b * NBOX * 4);
    uint32_t lds   = (uint32_t)(uintptr_t)(&s_box[tid]);
    uint32_t off   = idx * 16u;
    asm volatile("global_load_async_to_lds_b128 %0, %1, %2"
                 :: "v"(lds), "v"(off), "s"(bbase) : "memory");
    asm volatile("s_wait_asynccnt 0" ::: "memory");
  }
  __syncthreads();

  // ---- 4a) suppression bit-matrix: row tid vs all j ----
  {
    float4 mb = s_box[tid];
    float my_area = fmaxf(mb.z - mb.x, 0.f) * fmaxf(mb.w - mb.y, 0.f);
    for (int w = 0; w < 8; ++w) {
      uint32_t bits = 0;
      for (int t = 0; t < 32; ++t) {
        int j = w * 32 + t;
        float4 ob = s_box[j];
        float oarea = fmaxf(ob.z - ob.x, 0.f) * fmaxf(ob.w - ob.y, 0.f);
        float iy1 = fmaxf(mb.x, ob.x), ix1 = fmaxf(mb.y, ob.y);
        float iy2 = fminf(mb.z, ob.z), ix2 = fminf(mb.w, ob.w);
        float inter = fmaxf(iy2 - iy1, 0.f) * fmaxf(ix2 - ix1, 0.f);
        float uni   = my_area + oarea - inter;
        float iou   = (uni > 0.f) ? (inter / uni) : 0.f;
        bits |= (iou > 0.5f ? 1u : 0u) << t;
      }
      s_sup[tid * 8 + w] = bits;
    }
  }
  __syncthreads();

  // ---- 4b/5) greedy keep (exact early-exit at 8), emit slots ----
  if (tid == 0) {
    uint32_t keepw[8] = {0, 0, 0, 0, 0, 0, 0, 0};
    int kept = 0;
    const int slotBase = (b * NCLS + c) * MAXPC;
    for (int i = 0; i < KCAND && kept < MAXPC; ++i) {
      unsigned long long key = s_keys[i];
      float sc = __uint_as_float((uint32_t)(key >> 32));
      if (!(sc > 0.5f)) break;  // sorted descending; rest are below threshold
      bool sup = false;
      for (int w = 0; w < 8; ++w)
        if (keepw[w] & s_sup[i * 8 + w]) sup = true;
      if (!sup) {
        keepw[i >> 5] |= 1u << (i & 31);
        slot_score[slotBase + kept] = sc;
        slot_ti[slotBase + kept]    = c * KCAND + i;
        slot_box[slotBase + kept]   = s_box[i];
        ++kept;
      }
    }
    for (; kept < MAXPC; ++kept) {
      slot_score[slotBase + kept] = 0.f;
      slot_ti[slotBase + kept]    = 0;
      slot_box[slotBase + kept]   = make_float4(0.f, 0.f, 0.f, 0.f);
    }
  }
}

// ---------------------------------------------------------------------------
// Kernel B: one workgroup per batch — global top-8 over the 640 kept slots,
// tie-break by lower flat index ti (matches lax.top_k), emit final outputs.
// Output layout (floats): boxes[8*8*4] | scores[8*8] | classes[8*8] | valid[8]
// ---------------------------------------------------------------------------
__global__ __launch_bounds__(256) void nms_final_kernel(
    const float* __restrict__ slot_score, const int* __restrict__ slot_ti,
    const float4* __restrict__ slot_box, float* __restrict__ out)
{
  const int b   = blockIdx.x;
  const int tid = threadIdx.x;
  __shared__ unsigned long long t_keys[1024];
  __shared__ uint32_t           t_pay[1024];

  const int base = b * NCLS * MAXPC;  // 640 slots per batch
  for (int i = tid; i < 1024; i += 256) {
    unsigned long long k = 0ull;
    if (i < NCLS * MAXPC) {
      float sc = slot_score[base + i];
      if (sc > 0.f) {
        uint32_t ti = (uint32_t)slot_ti[base + i];
        k = ((unsigned long long)__float_as_uint(sc) << 32) | (uint32_t)(~ti);
      }
    }
    t_keys[i] = k;
    t_pay[i]  = (uint32_t)i;
  }
  __syncthreads();

  for (unsigned k = 2; k <= 1024; k <<= 1) {
    for (unsigned j = k >> 1; j > 0; j >>= 1) {
      for (unsigned zb = 0; zb < 1024; zb += 256) {
        unsigned i   = zb + tid;
        unsigned ixj = i ^ j;
        if (ixj > i) {
          unsigned long long a = t_keys[i], d = t_keys[ixj];
          bool up = ((i & k) == 0);
          if (up ? (a < d) : (a > d)) {
            t_keys[i] = d; t_keys[ixj] = a;
            uint32_t p = t_pay[i]; t_pay[i] = t_pay[ixj]; t_pay[ixj] = p;
          }
        }
      }
      __syncthreads();
    }
  }

  float* out_boxes   = out;
  float* out_scores  = out + NB * 8 * 4;
  float* out_classes = out + NB * 8 * 4 + NB * 8;
  float* out_valid   = out + NB * 8 * 4 + NB * 8 * 2;

  if (tid < 8) {
    unsigned long long k = t_keys[tid];
    float sc = __uint_as_float((uint32_t)(k >> 32));
    bool  m  = (k != 0ull) && (sc > 0.f);
    float4 bb = make_float4(0.f, 0.f, 0.f, 0.f);
    int cls = 0;
    if (m) {
      uint32_t pay = t_pay[tid];
      bb = slot_box[base + pay];
      bb.x = fminf(fmaxf(bb.x, 0.f), 1.f);
      bb.y = fminf(fmaxf(bb.y, 0.f), 1.f);
      bb.z = fminf(fmaxf(bb.z, 0.f), 1.f);
      bb.w = fminf(fmaxf(bb.w, 0.f), 1.f);
      cls = slot_ti[base + pay] / KCAND;
    }
    float* ob = out_boxes + (b * 8 + tid) * 4;
    ob[0] = bb.x; ob[1] = bb.y; ob[2] = bb.z; ob[3] = bb.w;
    out_scores[b * 8 + tid]  = m ? sc : 0.f;
    out_classes[b * 8 + tid] = (float)cls;
  }
  if (tid == 0) {
    int v = 0;
    for (int r = 0; r < 8; ++r) {
      unsigned long long k = t_keys[r];
      float sc = __uint_as_float((uint32_t)(k >> 32));
      if (k != 0ull && sc > 0.f) ++v;
    }
    out_valid[b] = (float)v;
  }
}

extern "C" void kernel_launch(void* const* d_in, const int* in_sizes, int n_in,
                              void* d_out, int out_size, void* d_ws, size_t ws_size,
                              hipStream_t stream) {
  const float* boxes  = (const float*)d_in[0];  // (8, 20000, 1, 4) f32
  const float* scores = (const float*)d_in[1];  // (8, 20000, 80)   f32

  // workspace layout: score[5120] f32 | ti[5120] i32 | box[5120] float4
  float*  slot_score = (float*)d_ws;
  int*    slot_ti    = (int*)((char*)d_ws + NB * NCLS * MAXPC * sizeof(float));
  float4* slot_box   = (float4*)((char*)d_ws + 2 * NB * NCLS * MAXPC * sizeof(float));

  nms_class_kernel<<<dim3(NCLS, NB), 256, 0, stream>>>(boxes, scores,
                                                       slot_score, slot_ti, slot_box);
  nms_final_kernel<<<dim3(NB), 256, 0, stream>>>(slot_score, slot_ti, slot_box,
                                                 (float*)d_out);
}